// AttentionModule_47665547051319
// MI455X (gfx1250) — compile-verified
//
#include <hip/hip_runtime.h>
#include <math.h>

#define NN 50000
#define EE 800000
#define FF 128
#define HH 8
#define DD 16

typedef __attribute__((ext_vector_type(2))) float v2f;
typedef __attribute__((ext_vector_type(8))) float v8f;

// ---------------------------------------------------------------------------
// Y = X @ W^T + b  ;  Y[n,j] = sum_k X[n,k] * W[j,k] + b[j]
// Block = 16 rows x 128 cols. 8 waves; wave w owns 16-col tile w.
// Uses V_WMMA_F32_16X16X4_F32, accumulating K=128 in 32 steps of 4.
// ---------------------------------------------------------------------------
__global__ void gemm_xwT_wmma(const float* __restrict__ X,
                              const float* __restrict__ W,
                              const float* __restrict__ b,
                              float* __restrict__ Y,
                              int nrows) {
  const int row0 = blockIdx.x * 16;
  if (row0 >= nrows) return;
  const int lane = threadIdx.x & 31;
  const int wave = threadIdx.x >> 5;       // 0..7 -> column tile
  const int col0 = wave * 16;
  const int half = lane >> 4;              // 0: K=k0,k0+1 ; 1: K=k0+2,k0+3
  const int lr   = lane & 15;              // A: row m ; B: col n

  const float* xrow = X + (size_t)(row0 + lr) * FF;   // A row (M = lr)
  const float* wrow = W + (size_t)(col0 + lr) * FF;   // B col (N = lr), W^T

  v8f acc = {};
#pragma unroll
  for (int k0 = 0; k0 < FF; k0 += 4) {
    const int ks = k0 + 2 * half;
    v2f a  = *(const v2f*)(xrow + ks);
    v2f bb = *(const v2f*)(wrow + ks);
    acc = __builtin_amdgcn_wmma_f32_16x16x4_f32(false, a, false, bb,
                                                (short)0, acc, false, false);
  }

  const float bias = b[col0 + lr];
#pragma unroll
  for (int r = 0; r < 8; ++r) {
    const int m = r + 8 * half;            // C/D layout: VGPR r -> row r (+8 for hi half)
    Y[(size_t)(row0 + m) * FF + col0 + lr] = acc[r] + bias;
  }
}

// ---------------------------------------------------------------------------
// Init: attn accumulator = 0, seg_max = -inf, seg_sum = 0
// ---------------------------------------------------------------------------
__global__ void init_ws(float* __restrict__ attn,
                        float* __restrict__ segmax,
                        float* __restrict__ segsum) {
  const int idx = blockIdx.x * blockDim.x + threadIdx.x;
  if (idx < NN * FF) attn[idx] = 0.0f;
  if (idx < NN * HH) {
    segmax[idx] = -__builtin_inff();
    segsum[idx] = 0.0f;
  }
}

__device__ __forceinline__ float edge_score(const float* __restrict__ q,
                                            const float* __restrict__ k,
                                            const float* __restrict__ mask,
                                            int e, int h, int s, int d) {
  const float4* qp = (const float4*)(q + (size_t)s * FF + h * DD);
  const float4* kp = (const float4*)(k + (size_t)d * FF + h * DD);
  float acc = 0.0f;
#pragma unroll
  for (int i = 0; i < 4; ++i) {
    float4 a = qp[i], bb = kp[i];
    acc += a.x * bb.x + a.y * bb.y + a.z * bb.z + a.w * bb.w;
  }
  return acc * 0.25f + mask[(size_t)e * HH + h];   // 1/sqrt(16) = 0.25
}

__device__ __forceinline__ void atomicMaxFloat(float* addr, float val) {
  if (val >= 0.0f)
    atomicMax((int*)addr, __float_as_int(val));
  else
    atomicMin((unsigned int*)addr, __float_as_uint(val));
}

// ---------------------------------------------------------------------------
// Pass 1: seg_max[src,h] = max over edges of score
// ---------------------------------------------------------------------------
__global__ void edge_max(const float* __restrict__ q, const float* __restrict__ k,
                         const int* __restrict__ ei, const float* __restrict__ mask,
                         float* __restrict__ segmax) {
  const int idx = blockIdx.x * blockDim.x + threadIdx.x;
  if (idx >= EE * HH) return;
  const int e = idx / HH, h = idx - e * HH;
  const int s = ei[e], d = ei[EE + e];
  const float sc = edge_score(q, k, mask, e, h, s, d);
  atomicMaxFloat(&segmax[(size_t)s * HH + h], sc);
}

// ---------------------------------------------------------------------------
// Pass 2: seg_sum[src,h] = sum exp(score - seg_max[src,h])
// ---------------------------------------------------------------------------
__global__ void edge_sum(const float* __restrict__ q, const float* __restrict__ k,
                         const int* __restrict__ ei, const float* __restrict__ mask,
                         const float* __restrict__ segmax,
                         float* __restrict__ segsum) {
  const int idx = blockIdx.x * blockDim.x + threadIdx.x;
  if (idx >= EE * HH) return;
  const int e = idx / HH, h = idx - e * HH;
  const int s = ei[e], d = ei[EE + e];
  const float sc = edge_score(q, k, mask, e, h, s, d);
  const float ex = __expf(sc - segmax[(size_t)s * HH + h]);
  atomicAdd(&segsum[(size_t)s * HH + h], ex);
}

// ---------------------------------------------------------------------------
// Pass 3: attn[dest, h*16+d] += v[src, h*16+d] * exp(score-max)/sum
// ---------------------------------------------------------------------------
__global__ void edge_scatter(const float* __restrict__ q, const float* __restrict__ k,
                             const float* __restrict__ v,
                             const int* __restrict__ ei, const float* __restrict__ mask,
                             const float* __restrict__ segmax,
                             const float* __restrict__ segsum,
                             float* __restrict__ attn) {
  const int idx = blockIdx.x * blockDim.x + threadIdx.x;
  if (idx >= EE * HH) return;
  const int e = idx / HH, h = idx - e * HH;
  const int s = ei[e], d = ei[EE + e];
  const float sc = edge_score(q, k, mask, e, h, s, d);
  const float w = __expf(sc - segmax[(size_t)s * HH + h]) /
                  segsum[(size_t)s * HH + h];
  const float* vp = v + (size_t)s * FF + h * DD;
  float* op = attn + (size_t)d * FF + h * DD;
#pragma unroll
  for (int i = 0; i < DD; ++i) {
    atomicAdd(&op[i], vp[i] * w);
  }
}

// ---------------------------------------------------------------------------
extern "C" void kernel_launch(void* const* d_in, const int* in_sizes, int n_in,
                              void* d_out, int out_size, void* d_ws, size_t ws_size,
                              hipStream_t stream) {
  const float* X    = (const float*)d_in[0];
  const int*   ei   = (const int*)  d_in[1];
  const float* mask = (const float*)d_in[2];
  const float* Wq   = (const float*)d_in[3];
  const float* bq   = (const float*)d_in[4];
  const float* Wk   = (const float*)d_in[5];
  const float* bk   = (const float*)d_in[6];
  const float* Wv   = (const float*)d_in[7];
  const float* bv   = (const float*)d_in[8];
  const float* Wo   = (const float*)d_in[9];
  const float* bo   = (const float*)d_in[10];
  float* out = (float*)d_out;

  // Workspace layout (floats): q | k | v | attn | segmax | segsum
  float* q      = (float*)d_ws;
  float* kk     = q    + (size_t)NN * FF;
  float* v      = kk   + (size_t)NN * FF;
  float* attn   = v    + (size_t)NN * FF;
  float* segmax = attn + (size_t)NN * FF;
  float* segsum = segmax + (size_t)NN * HH;

  const int rowTiles = NN / 16;                     // 50000 / 16 = 3125
  const dim3 gemmGrid(rowTiles), gemmBlock(256);

  gemm_xwT_wmma<<<gemmGrid, gemmBlock, 0, stream>>>(X, Wq, bq, q,  NN);
  gemm_xwT_wmma<<<gemmGrid, gemmBlock, 0, stream>>>(X, Wk, bk, kk, NN);
  gemm_xwT_wmma<<<gemmGrid, gemmBlock, 0, stream>>>(X, Wv, bv, v,  NN);

  const int initThreads = NN * FF;
  init_ws<<<(initThreads + 255) / 256, 256, 0, stream>>>(attn, segmax, segsum);

  const int ethreads = EE * HH;
  const int eblocks = (ethreads + 255) / 256;
  edge_max    <<<eblocks, 256, 0, stream>>>(q, kk, ei, mask, segmax);
  edge_sum    <<<eblocks, 256, 0, stream>>>(q, kk, ei, mask, segmax, segsum);
  edge_scatter<<<eblocks, 256, 0, stream>>>(q, kk, v, ei, mask, segmax, segsum, attn);

  gemm_xwT_wmma<<<gemmGrid, gemmBlock, 0, stream>>>(attn, Wo, bo, out, NN);
}